// AttentionControl_57019985822161
// MI455X (gfx1250) — compile-verified
//
#include <hip/hip_runtime.h>
#include <hip/hip_bf16.h>
#include <math.h>

typedef __attribute__((ext_vector_type(2))) float v2f;
typedef __attribute__((ext_vector_type(8))) float v8f;

#define Bsz   256
#define IMG   512
#define SB    32
#define TSB   3072
#define UNITS 256
#define HGN   128
#define GN    256

// ---------------------------------------------------------------------------
// Kernel 1: location head + hl (tiny K=2 GEMV). One thread per batch row.
// ---------------------------------------------------------------------------
__global__ __launch_bounds__(256) void loc_hl_kernel(
    const float* __restrict__ output, const float* __restrict__ noise,
    const float* __restrict__ Wm, const float* __restrict__ bm,
    const float* __restrict__ Ws, const float* __restrict__ bs,
    const float* __restrict__ Whl, const float* __restrict__ bhl,
    float* __restrict__ loc, float* __restrict__ hl)
{
    const int b = threadIdx.x;  // 0..255
    const float* orow = output + b * UNITS;
    float m0 = 0.f, m1 = 0.f, s0 = 0.f, s1 = 0.f;
    for (int k = 0; k < UNITS; ++k) {
        float o = orow[k];
        m0 = fmaf(o, Wm[k * 2 + 0], m0);
        m1 = fmaf(o, Wm[k * 2 + 1], m1);
        s0 = fmaf(o, Ws[k * 2 + 0], s0);
        s1 = fmaf(o, Ws[k * 2 + 1], s1);
    }
    float mean0 = fminf(fmaxf(m0 + bm[0], -1.f), 1.f);
    float mean1 = fminf(fmaxf(m1 + bm[1], -1.f), 1.f);
    float std0  = 1.f / (1.f + __expf(-(s0 + bs[0])));
    float std1  = 1.f / (1.f + __expf(-(s1 + bs[1])));
    float samp0 = fminf(fmaxf(mean0 + noise[b * 2 + 0] * std0, -1.f), 1.f);
    float samp1 = fminf(fmaxf(mean1 + noise[b * 2 + 1] * std1, -1.f), 1.f);
    float l0 = samp0 * 2.0f;  // PS = 2
    float l1 = samp1 * 2.0f;
    loc[b * 2 + 0] = l0;
    loc[b * 2 + 1] = l1;
    // hl = relu(loc @ Whl + bhl), Whl is [2,128] row-major
    float* hrow = hl + b * HGN;
    for (int j = 0; j < HGN; ++j) {
        float h = fmaf(l0, Whl[j], fmaf(l1, Whl[HGN + j], bhl[j]));
        hrow[j] = h > 0.f ? h : 0.f;
    }
}

// ---------------------------------------------------------------------------
// Kernel 2: glimpse sensor. One block per batch image; zero-pad emulated by
// bounds check; linear/no-antialias resize at x2/x4 == 2x2 box average.
// ---------------------------------------------------------------------------
__device__ __forceinline__ float px(const float* im, int y, int x) {
    return (y >= 0 && y < IMG && x >= 0 && x < IMG) ? im[y * IMG + x] : 0.f;
}

__global__ __launch_bounds__(256) void glimpse_kernel(
    const float* __restrict__ img, const float* __restrict__ loc,
    float* __restrict__ glimpses)
{
    const int b = blockIdx.x;
    const float l0 = loc[b * 2 + 0];
    const float l1 = loc[b * 2 + 1];
    // jnp.round == round-half-even == rintf in default mode
    const int lp0 = (int)rintf((l0 + 1.f) * 0.5f * (float)IMG);
    const int lp1 = (int)rintf((l1 + 1.f) * 0.5f * (float)IMG);
    // dynamic_slice clamped starts in padded coords (pad offset = 128)
    const int s0y = min(max(lp0 + 112, 0), 736), s0x = min(max(lp1 + 112, 0), 736);
    const int s1y = min(max(lp0 +  96, 0), 704), s1x = min(max(lp1 +  96, 0), 704);
    const int s2y = min(max(lp0 +  64, 0), 640), s2x = min(max(lp1 +  64, 0), 640);
    const float* im = img + (size_t)b * IMG * IMG;
    float* g = glimpses + (size_t)b * TSB;

    for (int p = threadIdx.x; p < SB * SB; p += 256) {
        const int i = p >> 5, j = p & 31;
        // depth 0: direct 32x32 crop
        g[p] = px(im, s0y - 128 + i, s0x - 128 + j);
        // depth 1: 64->32, sample pos 2i+0.5 -> avg rows {2i,2i+1}
        {
            int y = s1y - 128 + 2 * i, x = s1x - 128 + 2 * j;
            g[1024 + p] = 0.25f * (px(im, y, x) + px(im, y, x + 1) +
                                   px(im, y + 1, x) + px(im, y + 1, x + 1));
        }
        // depth 2: 128->32, sample pos 4i+1.5 -> avg rows {4i+1,4i+2}
        {
            int y = s2y - 128 + 4 * i + 1, x = s2x - 128 + 4 * j + 1;
            g[2048 + p] = 0.25f * (px(im, y, x) + px(im, y, x + 1) +
                                   px(im, y + 1, x) + px(im, y + 1, x + 1));
        }
    }
}

// ---------------------------------------------------------------------------
// WMMA fragment helpers (V_WMMA_F32_16X16X4_F32 layouts, ISA 7.12.2):
//  A 16x4: lane<16 row M=lane holds {K, K+1}; lane>=16 same rows hold {K+2,K+3}
//  B 4x16: lane<16 col N=lane holds {K, K+1}; lane>=16 same cols hold {K+2,K+3}
//  C/D 16x16: lane&15 = N, VGPR v = row v + 8*(lane>=16)
// ---------------------------------------------------------------------------

// Kernel 3: hg = relu(glimpses[256,3072] @ Whg[3072,128] + bhg)
// grid = 16 M-tiles; 8 waves/block = 8 N-tiles (covers all 128 cols).
__global__ __launch_bounds__(256) void hg_gemm_kernel(
    const float* __restrict__ G, const float* __restrict__ W,
    const float* __restrict__ bias, float* __restrict__ hg)
{
    const int lane = threadIdx.x & 31;
    const int wave = threadIdx.x >> 5;
    const int half = lane >> 4;        // 0 | 1
    const int idx  = lane & 15;
    const int mBase = blockIdx.x * 16;
    const int nBase = wave * 16;
    const float* Arow = G + (size_t)(mBase + idx) * TSB;   // A row m = idx
    const float* Bcol = W + nBase + idx;                   // B col n = idx, row stride 128

    __builtin_prefetch(Arow, 0, 3);              // global_prefetch_b8
    __builtin_prefetch(Bcol, 0, 3);

    v8f c = {};
    for (int k = 0; k < TSB; k += 4) {
        const int ka = k + 2 * half;
        v2f a, b;
        a.x = Arow[ka];
        a.y = Arow[ka + 1];
        b.x = Bcol[(size_t)ka * HGN];
        b.y = Bcol[(size_t)(ka + 1) * HGN];
        c = __builtin_amdgcn_wmma_f32_16x16x4_f32(false, a, false, b,
                                                  (short)0, c, false, false);
    }
    const float bn = bias[nBase + idx];
#pragma unroll
    for (int v = 0; v < 8; ++v) {
        const int m = mBase + v + 8 * half;
        float val = c[v] + bn;
        hg[m * HGN + nBase + idx] = val > 0.f ? val : 0.f;
    }
}

// Kernel 4: g = relu(hg[256,128]@Wg1[128,256] + hl[256,128]@Wg2[128,256] + b1+b2)
// grid = 16 M-tiles * 2 N-groups; 8 waves/block cover 128 cols each group.
__global__ __launch_bounds__(256) void out_gemm_kernel(
    const float* __restrict__ hg, const float* __restrict__ hl,
    const float* __restrict__ Wg1, const float* __restrict__ Wg2,
    const float* __restrict__ bg1, const float* __restrict__ bg2,
    float* __restrict__ out)
{
    const int lane = threadIdx.x & 31;
    const int wave = threadIdx.x >> 5;
    const int half = lane >> 4;
    const int idx  = lane & 15;
    const int mBase = (blockIdx.x >> 1) * 16;
    const int nBase = (blockIdx.x & 1) * 128 + wave * 16;

    const float* A1 = hg + (size_t)(mBase + idx) * HGN;
    const float* A2 = hl + (size_t)(mBase + idx) * HGN;
    const float* B1 = Wg1 + nBase + idx;   // row stride 256
    const float* B2 = Wg2 + nBase + idx;

    __builtin_prefetch(A1, 0, 3);
    __builtin_prefetch(B1, 0, 3);

    v8f c = {};
    for (int k = 0; k < HGN; k += 4) {
        const int ka = k + 2 * half;
        v2f a, b;
        a.x = A1[ka];            a.y = A1[ka + 1];
        b.x = B1[(size_t)ka * GN]; b.y = B1[(size_t)(ka + 1) * GN];
        c = __builtin_amdgcn_wmma_f32_16x16x4_f32(false, a, false, b,
                                                  (short)0, c, false, false);
    }
    for (int k = 0; k < HGN; k += 4) {
        const int ka = k + 2 * half;
        v2f a, b;
        a.x = A2[ka];            a.y = A2[ka + 1];
        b.x = B2[(size_t)ka * GN]; b.y = B2[(size_t)(ka + 1) * GN];
        c = __builtin_amdgcn_wmma_f32_16x16x4_f32(false, a, false, b,
                                                  (short)0, c, false, false);
    }
    const float bn = bg1[nBase + idx] + bg2[nBase + idx];
#pragma unroll
    for (int v = 0; v < 8; ++v) {
        const int m = mBase + v + 8 * half;
        float val = c[v] + bn;
        out[m * GN + nBase + idx] = val > 0.f ? val : 0.f;
    }
}

// ---------------------------------------------------------------------------
extern "C" void kernel_launch(void* const* d_in, const int* in_sizes, int n_in,
                              void* d_out, int out_size, void* d_ws, size_t ws_size,
                              hipStream_t stream) {
    const float* inputs = (const float*)d_in[0];
    const float* output = (const float*)d_in[1];
    const float* noise  = (const float*)d_in[2];
    const float* Wm  = (const float*)d_in[3];
    const float* bm  = (const float*)d_in[4];
    const float* Ws  = (const float*)d_in[5];
    const float* bs  = (const float*)d_in[6];
    const float* Whg = (const float*)d_in[7];
    const float* bhg = (const float*)d_in[8];
    const float* Whl = (const float*)d_in[9];
    const float* bhl = (const float*)d_in[10];
    const float* Wg1 = (const float*)d_in[11];
    const float* bg1 = (const float*)d_in[12];
    const float* Wg2 = (const float*)d_in[13];
    const float* bg2 = (const float*)d_in[14];

    float* ws       = (float*)d_ws;
    float* loc      = ws;                       // 256*2
    float* hl       = loc + Bsz * 2;            // 256*128
    float* glimpses = hl + Bsz * HGN;           // 256*3072
    float* hg       = glimpses + Bsz * TSB;     // 256*128

    loc_hl_kernel<<<1, 256, 0, stream>>>(output, noise, Wm, bm, Ws, bs,
                                         Whl, bhl, loc, hl);
    glimpse_kernel<<<Bsz, 256, 0, stream>>>(inputs, loc, glimpses);
    hg_gemm_kernel<<<16, 256, 0, stream>>>(glimpses, Whg, bhg, hg);
    out_gemm_kernel<<<32, 256, 0, stream>>>(hg, hl, Wg1, Wg2, bg1, bg2,
                                            (float*)d_out);
}